// TinyLSTM_68753836474968
// MI455X (gfx1250) — compile-verified
//
#include <hip/hip_runtime.h>
#include <hip/hip_bf16.h>

typedef __attribute__((ext_vector_type(2))) float v2f;
typedef __attribute__((ext_vector_type(4))) float v4f;
typedef __attribute__((ext_vector_type(8))) float v8f;

#define HDIM 32   // hidden size
#define NT   8    // N tiles: 4*H / 16
#define KG   8    // K groups: H / 4

// ---- nonlinearities -------------------------------------------------------
#if __has_builtin(__builtin_amdgcn_tanhf)
// CDNA5 has native v_tanh_f32 (TRANS pipe, co-executes with XDL WMMA)
__device__ __forceinline__ float ftanh(float x) {
    return __builtin_amdgcn_tanhf(x);
}
__device__ __forceinline__ float fsigmoid(float x) {
    // sigmoid(x) = 0.5 * tanh(x/2) + 0.5
    return __builtin_fmaf(0.5f, __builtin_amdgcn_tanhf(0.5f * x), 0.5f);
}
#else
// fallback: exp2 + rcp hardware ops
__device__ __forceinline__ float fsigmoid(float x) {
    return __builtin_amdgcn_rcpf(1.0f + __builtin_amdgcn_exp2f(x * -1.4426950408889634f));
}
__device__ __forceinline__ float ftanh(float x) {
    return 2.0f * __builtin_amdgcn_rcpf(1.0f + __builtin_amdgcn_exp2f(x * -2.8853900817779268f)) - 1.0f;
}
#endif

__global__ __launch_bounds__(32, 1)
void tiny_lstm_wmma_f32(const float* __restrict__ x,      // [B, T, 1]
                        const float* __restrict__ W_ih,   // [4H, 1]
                        const float* __restrict__ W_hh,   // [4H, H] row-major
                        const float* __restrict__ b_ih,   // [4H]
                        const float* __restrict__ b_hh,   // [4H]
                        const float* __restrict__ fc_w,   // [1, H]
                        const float* __restrict__ fc_b,   // [1]
                        float* __restrict__ out,          // [B]
                        int T)
{
    // h staging buffer: layout [hidden j][local batch m] (m fastest, 16 wide)
    __shared__ float h_lds[HDIM * 16];

    const int lane = threadIdx.x & 31;
    const int l16  = lane & 15;
    const int half = lane >> 4;              // 0: lanes 0-15, 1: lanes 16-31
    const int m0   = blockIdx.x * 16;        // global batch tile base

    // ---------------------------------------------------------------
    // Resident B operand: B[k][n] = W_hh[n][k]  (gates = h @ W_hh^T)
    // B-frag (4x16, KxN): lane -> n = n0+l16; VGPR0 = row k0+2*half,
    // VGPR1 = row k0+2*half+1.  Loaded as one float2 per fragment.
    // ---------------------------------------------------------------
    v2f Bw[KG][NT];
    #pragma unroll
    for (int t8 = 0; t8 < NT; ++t8) {
        const int n = t8 * 16 + l16;
        #pragma unroll
        for (int kg = 0; kg < KG; ++kg) {
            const int k = kg * 4 + 2 * half;
            Bw[kg][t8] = *(const v2f*)(W_hh + n * HDIM + k); // {W_hh[n][k], W_hh[n][k+1]}
        }
    }
    // Extra K-group (K-extension): B rows 32 = W_ih, 33 = b_ih+b_hh, 34/35 = 0
    v2f Bx[NT];
    #pragma unroll
    for (int t8 = 0; t8 < NT; ++t8) {
        const int n = t8 * 16 + l16;
        const float w  = W_ih[n];
        const float bb = b_ih[n] + b_hh[n];
        Bx[t8].x = half ? 0.0f : w;   // row 32 (half0) / row 34 = 0 (half1)
        Bx[t8].y = half ? 0.0f : bb;  // row 33 (half0) / row 35 = 0 (half1)
    }

    // h0 = 0
    #pragma unroll
    for (int i = 0; i < 16; ++i) h_lds[i * 32 + lane] = 0.0f;

    // cell state in C-layout: frag0 = hidden 0-15 (j = l16), frag1 = hidden 16-31
    // element (v, lane): batch M = v + 8*half, hidden j = l16 (+16)
    v8f c0 = {}, c1 = {};

    float xv = x[(size_t)(m0 + l16) * T];    // prefetched x_t for batch m0+l16

    for (int t = 0; t < T; ++t) {
        const float xnext = (t + 1 < T) ? x[(size_t)(m0 + l16) * T + t + 1] : 0.0f;

        // A extension fragment: A[m][32]=x_m, A[m][33]=1, A[m][34..35]=0
        v2f Ax;
        Ax.x = half ? 0.0f : xv;
        Ax.y = half ? 0.0f : 1.0f;

        // C init = x * W_ih + bias via the extension WMMA (C input = 0)
        v8f C[NT];
        #pragma unroll
        for (int t8 = 0; t8 < NT; ++t8) {
            v8f z = {};
            C[t8] = __builtin_amdgcn_wmma_f32_16x16x4_f32(
                        false, Ax, false, Bx[t8], (short)0, z, false, false);
        }

        // A fragments of h from LDS (stores from previous iteration are
        // in-wave ordered before these reads; fence keeps program order)
        asm volatile("" ::: "memory");
        v2f A[KG];
        #pragma unroll
        for (int kg = 0; kg < KG; ++kg) {
            const int k = kg * 4 + 2 * half;
            A[kg].x = h_lds[k * 16 + l16];        // h[M=l16][k]
            A[kg].y = h_lds[(k + 1) * 16 + l16];  // h[M=l16][k+1]
        }

        // gates += h @ W_hh^T : 64 WMMAs
        #pragma unroll
        for (int kg = 0; kg < KG; ++kg) {
            #pragma unroll
            for (int t8 = 0; t8 < NT; ++t8) {
                C[t8] = __builtin_amdgcn_wmma_f32_16x16x4_f32(
                            false, A[kg], false, Bw[kg][t8], (short)0, C[t8], false, false);
            }
        }

        // elementwise LSTM cell update; tiles: 0,1=i  2,3=f  4,5=g  6,7=o
        // write h straight to LDS (scalar stores keep VGPR pressure down)
        asm volatile("" ::: "memory");
        #pragma unroll
        for (int v = 0; v < 8; ++v) {
            {   // hidden j = l16
                const float ig = fsigmoid(C[0][v]);
                const float fg = fsigmoid(C[2][v]);
                const float gg = ftanh  (C[4][v]);
                const float og = fsigmoid(C[6][v]);
                const float cc = fg * c0[v] + ig * gg;
                c0[v] = cc;
                h_lds[l16 * 16 + v + 8 * half] = og * ftanh(cc);
            }
            {   // hidden j = 16 + l16
                const float ig = fsigmoid(C[1][v]);
                const float fg = fsigmoid(C[3][v]);
                const float gg = ftanh  (C[5][v]);
                const float og = fsigmoid(C[7][v]);
                const float cc = fg * c1[v] + ig * gg;
                c1[v] = cc;
                h_lds[(16 + l16) * 16 + v + 8 * half] = og * ftanh(cc);
            }
        }

        xv = xnext;
    }

    // Epilogue: out[b] = h_last[b] . fc_w + fc_b   (h_last already in LDS)
    asm volatile("" ::: "memory");
    if (lane < 16) {
        float acc = fc_b[0];
        #pragma unroll
        for (int j = 0; j < HDIM; ++j)
            acc += h_lds[j * 16 + lane] * fc_w[j];
        out[m0 + lane] = acc;
    }
}

extern "C" void kernel_launch(void* const* d_in, const int* in_sizes, int n_in,
                              void* d_out, int out_size, void* d_ws, size_t ws_size,
                              hipStream_t stream) {
    const float* x    = (const float*)d_in[0];
    const float* W_ih = (const float*)d_in[1];
    const float* W_hh = (const float*)d_in[2];
    const float* b_ih = (const float*)d_in[3];
    const float* b_hh = (const float*)d_in[4];
    const float* fc_w = (const float*)d_in[5];
    const float* fc_b = (const float*)d_in[6];
    float* out = (float*)d_out;

    const int B = out_size;                 // 8192
    const int T = in_sizes[0] / B;          // 512

    dim3 grid(B / 16), block(32);           // one wave32 per block, 16 batches/wave
    tiny_lstm_wmma_f32<<<grid, block, 0, stream>>>(x, W_ih, W_hh, b_ih, b_hh,
                                                   fc_w, fc_b, out, T);
}